// MambaBlock_34394098106757
// MI455X (gfx1250) — compile-verified
//
#include <hip/hip_runtime.h>
#include <hip/hip_bf16.h>
#include <math.h>

typedef __attribute__((ext_vector_type(2))) float v2f;
typedef __attribute__((ext_vector_type(8))) float v8f;

#define DIM      1024
#define D_STATE  16
#define D_CONV   4
#define D_INNER  2048
#define DT_RANK  64
#define BATCH    2
#define SEQ      1024
#define ROWS     (BATCH * SEQ)          // 2048
#define PROJW    (DT_RANK + 2*D_STATE)  // 96
#define CH       16                     // scan chunks along L
#define CLEN     (SEQ / CH)             // 64
#define TOT      (BATCH * CH * D_INNER) // 65536 chunk-threads

// ---------------------------------------------------------------------------
// 64x64-per-wave fp32 WMMA GEMM:  C[m][n] = sum_k A[m*lda+k] * Bt[n*ldb+k]
// 16 accumulator tiles (128 VGPRs); per K-step-of-4: 8 float2 loads feed
// 16 v_wmma_f32_16x16x4_f32 -> 16 FLOP/B of global/L2 traffic.
// A operand layout (ISA 7.12.2): lanes 0-15 -> M=lane, K={0,1};
// lanes 16-31 -> M=lane-16, K={2,3}; B mirrors with N on the lane axis.
// Requires M,N % 64 == 0, K % 4 == 0, rows 8B-aligned.
// ---------------------------------------------------------------------------
__global__ __launch_bounds__(32)
void gemm_wmma_f32_64(const float* __restrict__ A, const float* __restrict__ Bt,
                      float* __restrict__ C, int K, int lda, int ldb, int ldc) {
    const int tileN = blockIdx.x * 64;
    const int tileM = blockIdx.y * 64;
    const int lane  = threadIdx.x;
    const int r     = lane & 15;
    const int hh    = lane >> 4;

    const float* aptr = A  + (size_t)(tileM + r) * lda + 2 * hh;
    const float* bptr = Bt + (size_t)(tileN + r) * ldb + 2 * hh;

    v8f acc[4][4] = {};
    for (int k = 0; k < K; k += 4) {
        v2f a[4], b[4];
#pragma unroll
        for (int i = 0; i < 4; ++i) a[i] = *(const v2f*)(aptr + (size_t)(16*i)*lda + k);
#pragma unroll
        for (int j = 0; j < 4; ++j) b[j] = *(const v2f*)(bptr + (size_t)(16*j)*ldb + k);
#pragma unroll
        for (int i = 0; i < 4; ++i)
#pragma unroll
            for (int j = 0; j < 4; ++j)
                acc[i][j] = __builtin_amdgcn_wmma_f32_16x16x4_f32(
                    false, a[i], false, b[j], (short)0, acc[i][j], false, false);
    }
#pragma unroll
    for (int i = 0; i < 4; ++i)
#pragma unroll
        for (int j = 0; j < 4; ++j) {
            float* cptr = C + (size_t)(tileM + 16*i + 8*hh) * ldc + tileN + 16*j + r;
#pragma unroll
            for (int t = 0; t < 8; ++t) cptr[(size_t)t * ldc] = acc[i][j][t];
        }
}

// 16x16-per-wave variant for N not divisible by 64 (proj GEMM, N=96).
__global__ __launch_bounds__(32)
void gemm_wmma_f32_16(const float* __restrict__ A, const float* __restrict__ Bt,
                      float* __restrict__ C, int K, int lda, int ldb, int ldc) {
    const int tileN = blockIdx.x * 16;
    const int tileM = blockIdx.y * 16;
    const int lane  = threadIdx.x;
    const int r     = lane & 15;
    const int hh    = lane >> 4;
    const float* aptr = A  + (size_t)(tileM + r) * lda + 2 * hh;
    const float* bptr = Bt + (size_t)(tileN + r) * ldb + 2 * hh;
    v8f acc = {};
    for (int k = 0; k < K; k += 4) {
        v2f a = *(const v2f*)(aptr + k);
        v2f b = *(const v2f*)(bptr + k);
        acc = __builtin_amdgcn_wmma_f32_16x16x4_f32(false, a, false, b,
                                                    (short)0, acc, false, false);
    }
    float* cptr = C + (size_t)(tileM + 8*hh) * ldc + tileN + r;
#pragma unroll
    for (int i = 0; i < 8; ++i) cptr[(size_t)i * ldc] = acc[i];
}

__device__ __forceinline__ float silu_f(float v)     { return v / (1.0f + __expf(-v)); }
__device__ __forceinline__ float softplus_f(float v) { return (v > 20.0f) ? v : log1pf(__expf(v)); }

// ---------------------------------------------------------------------------
// Causal depthwise conv1d (k=4, left pad 3, per batch) + bias + SiLU.
// ---------------------------------------------------------------------------
__global__ __launch_bounds__(256)
void conv_silu_kernel(const float* __restrict__ xz, const float* __restrict__ Wc,
                      const float* __restrict__ bc, float* __restrict__ xs) {
    int idx = blockIdx.x * blockDim.x + threadIdx.x;
    if (idx >= ROWS * D_INNER) return;
    int d   = idx % D_INNER;
    int row = idx / D_INNER;
    int l   = row % SEQ;
    float acc = bc[d];
#pragma unroll
    for (int j = 0; j < D_CONV; ++j) {
        int ls = l - (D_CONV - 1) + j;
        if (ls >= 0)
            acc += xz[(size_t)(row - (D_CONV - 1) + j) * (2 * D_INNER) + d] * Wc[d * D_CONV + j];
    }
    xs[idx] = silu_f(acc);
}

// ---------------------------------------------------------------------------
// Chunked selective scan.  Recurrence h <- exp(dt*A)*h + dt*B*x is linear
// diagonal, so chunk c maps h_in -> exp(A*S_c)*h_in + q_c with S_c = sum(dt).
// Pass1: per (b,chunk,d) scan from h=0 -> S, q[16].     (TOT threads)
// Pass2: per (b,d) serial combine over 16 chunks -> h_init per chunk.
// Pass3: per (b,chunk,d) replay with h_init, emit gated y in place over xs.
// idx decode: d fastest => dt/xs/z coalesced, proj row broadcast per wave.
// ---------------------------------------------------------------------------
__global__ __launch_bounds__(256)
void scan_pass1(const float* __restrict__ dtraw, const float* __restrict__ b_dt,
                const float* __restrict__ proj,  const float* __restrict__ A_log,
                const float* __restrict__ xs,
                float* __restrict__ Sbuf, float* __restrict__ qbuf) {
    int idx = blockIdx.x * blockDim.x + threadIdx.x;   // 0..TOT-1
    int d  = idx % D_INNER;
    int bc = idx / D_INNER;
    int c  = bc % CH;
    int b  = bc / CH;

    float Ac[D_STATE];
#pragma unroll
    for (int n = 0; n < D_STATE; ++n) Ac[n] = -__expf(A_log[d * D_STATE + n]);
    float h[D_STATE];
#pragma unroll
    for (int n = 0; n < D_STATE; ++n) h[n] = 0.0f;

    const float bdt = b_dt[d];
    float S = 0.0f;
    const int l0 = c * CLEN;
    for (int l = l0; l < l0 + CLEN; ++l) {
        size_t row = (size_t)b * SEQ + l;
        float dtv = softplus_f(dtraw[row * D_INNER + d] + bdt);
        S += dtv;
        float xv = xs[row * D_INNER + d];
        const float* pr = proj + row * PROJW;
#pragma unroll
        for (int n = 0; n < D_STATE; ++n)
            h[n] = h[n] * __expf(dtv * Ac[n]) + dtv * pr[DT_RANK + n] * xv;
    }
    Sbuf[idx] = S;
#pragma unroll
    for (int n = 0; n < D_STATE; ++n) qbuf[(size_t)n * TOT + idx] = h[n];
}

__global__ __launch_bounds__(256)
void scan_pass2(const float* __restrict__ Sbuf, const float* __restrict__ qbuf,
                const float* __restrict__ A_log, float* __restrict__ hinit) {
    int idx = blockIdx.x * blockDim.x + threadIdx.x;   // 0..BATCH*D_INNER-1
    int d = idx % D_INNER;
    int b = idx / D_INNER;
    float Ac[D_STATE];
#pragma unroll
    for (int n = 0; n < D_STATE; ++n) Ac[n] = -__expf(A_log[d * D_STATE + n]);
    float h[D_STATE];
#pragma unroll
    for (int n = 0; n < D_STATE; ++n) h[n] = 0.0f;

    for (int c = 0; c < CH; ++c) {
        size_t ic = ((size_t)(b * CH + c)) * D_INNER + d;
#pragma unroll
        for (int n = 0; n < D_STATE; ++n) hinit[(size_t)n * TOT + ic] = h[n];
        float S = Sbuf[ic];
#pragma unroll
        for (int n = 0; n < D_STATE; ++n)
            h[n] = h[n] * __expf(Ac[n] * S) + qbuf[(size_t)n * TOT + ic];
    }
}

__global__ __launch_bounds__(256)
void scan_pass3(const float* __restrict__ dtraw, const float* __restrict__ b_dt,
                const float* __restrict__ proj,  const float* __restrict__ xz,
                const float* __restrict__ A_log, const float* __restrict__ D_param,
                const float* __restrict__ hinit, float* __restrict__ xs) {
    int idx = blockIdx.x * blockDim.x + threadIdx.x;   // 0..TOT-1
    int d  = idx % D_INNER;
    int bc = idx / D_INNER;
    int c  = bc % CH;
    int b  = bc / CH;

    float Ac[D_STATE];
#pragma unroll
    for (int n = 0; n < D_STATE; ++n) Ac[n] = -__expf(A_log[d * D_STATE + n]);
    float h[D_STATE];
#pragma unroll
    for (int n = 0; n < D_STATE; ++n) h[n] = hinit[(size_t)n * TOT + idx];

    const float Dp  = D_param[d];
    const float bdt = b_dt[d];
    const int l0 = c * CLEN;
    for (int l = l0; l < l0 + CLEN; ++l) {
        size_t row = (size_t)b * SEQ + l;
        float dtv = softplus_f(dtraw[row * D_INNER + d] + bdt);
        float xv  = xs[row * D_INNER + d];
        const float* pr = proj + row * PROJW;
        float y = 0.0f;
#pragma unroll
        for (int n = 0; n < D_STATE; ++n) {
            h[n] = h[n] * __expf(dtv * Ac[n]) + dtv * pr[DT_RANK + n] * xv;
            y += h[n] * pr[DT_RANK + D_STATE + n];
        }
        y += Dp * xv;
        float zv = xz[row * (2 * D_INNER) + D_INNER + d];
        y *= silu_f(zv);
        xs[row * D_INNER + d] = y;
    }
}

// ---------------------------------------------------------------------------
extern "C" void kernel_launch(void* const* d_in, const int* in_sizes, int n_in,
                              void* d_out, int out_size, void* d_ws, size_t ws_size,
                              hipStream_t stream) {
    const float* x       = (const float*)d_in[0];
    const float* W_in    = (const float*)d_in[1];
    const float* W_conv  = (const float*)d_in[2];
    const float* b_conv  = (const float*)d_in[3];
    const float* W_xproj = (const float*)d_in[4];
    const float* W_dt    = (const float*)d_in[5];
    const float* b_dt    = (const float*)d_in[6];
    const float* A_log   = (const float*)d_in[7];
    const float* D_param = (const float*)d_in[8];
    const float* W_out   = (const float*)d_in[9];
    float* out = (float*)d_out;

    // Workspace layout (floats)
    float* xz    = (float*)d_ws;                           // ROWS * 4096
    float* xs    = xz    + (size_t)ROWS * 2 * D_INNER;     // ROWS * 2048 (y in-place)
    float* proj  = xs    + (size_t)ROWS * D_INNER;         // ROWS * 96
    float* dtraw = proj  + (size_t)ROWS * PROJW;           // ROWS * 2048
    float* Sbuf  = dtraw + (size_t)ROWS * D_INNER;         // TOT
    float* qbuf  = Sbuf  + (size_t)TOT;                    // 16 * TOT
    float* hinit = qbuf  + (size_t)D_STATE * TOT;          // 16 * TOT

    dim3 wave(32);

    // 1) xz = x @ W_in^T : (2048x1024)x(4096x1024)^T -> (2048x4096)
    gemm_wmma_f32_64<<<dim3((2 * D_INNER) / 64, ROWS / 64), wave, 0, stream>>>(
        x, W_in, xz, DIM, DIM, DIM, 2 * D_INNER);

    // 2) conv + SiLU -> xs
    int tot = ROWS * D_INNER;
    conv_silu_kernel<<<(tot + 255) / 256, 256, 0, stream>>>(xz, W_conv, b_conv, xs);

    // 3) proj = xs @ W_xproj^T -> (2048x96)   (N=96 -> 16x16 kernel)
    gemm_wmma_f32_16<<<dim3(PROJW / 16, ROWS / 16), wave, 0, stream>>>(
        xs, W_xproj, proj, D_INNER, D_INNER, D_INNER, PROJW);

    // 4) dtraw = proj[:, :64] @ W_dt^T -> (2048x2048)
    gemm_wmma_f32_64<<<dim3(D_INNER / 64, ROWS / 64), wave, 0, stream>>>(
        proj, W_dt, dtraw, DT_RANK, PROJW, DT_RANK, D_INNER);

    // 5) chunked selective scan (softplus + recurrence + SiLU gate), y -> xs
    scan_pass1<<<TOT / 256, 256, 0, stream>>>(dtraw, b_dt, proj, A_log, xs, Sbuf, qbuf);
    scan_pass2<<<(BATCH * D_INNER) / 256, 256, 0, stream>>>(Sbuf, qbuf, A_log, hinit);
    scan_pass3<<<TOT / 256, 256, 0, stream>>>(dtraw, b_dt, proj, xz, A_log, D_param, hinit, xs);

    // 6) out = y @ W_out^T -> (2048x1024)
    gemm_wmma_f32_64<<<dim3(DIM / 64, ROWS / 64), wave, 0, stream>>>(
        xs, W_out, out, D_INNER, D_INNER, D_INNER, DIM);
}